// Model_79688823210138
// MI455X (gfx1250) — compile-verified
//
#include <hip/hip_runtime.h>
#include <math.h>

typedef __attribute__((ext_vector_type(2))) float v2f;
typedef __attribute__((ext_vector_type(8))) float v8f;

namespace {
constexpr int kB  = 4096;
constexpr int kT  = 128;
constexpr int kD  = 128;
constexpr int kH  = 150;
constexpr int kO  = 10;
constexpr int kHP = 160;        // hidden padded to 10 tiles of 16
constexpr int kNT = kHP / 16;   // 10 N tiles total
constexpr int kNTW = kNT / 2;   // 5 N tiles per wave (2-way N split)
constexpr int kKTX = kD / 4;    // 32 K-tiles (f32 WMMA K=4) for the x part
constexpr int kKTH = kHP / 4;   // 40 K-tiles for the h part
constexpr int kRows = 32;       // batch rows per block (2 row tiles x 2 N halves = 4 waves)
constexpr int kThreads = 128;
constexpr int kXStride = 132;   // padded LDS strides -> conflict-free A-frag ds_load_b64
constexpr int kHStride = 164;
constexpr int kXSlots = kRows * (kD / 4);        // 1024 float4 slots per x tile
constexpr int kXPerThr = kXSlots / kThreads;     // 8 float4 per thread
}

__device__ __forceinline__ float fast_tanh(float x) {
  // tanh(x) = 1 - 2/(exp(2x)+1); exp overflow -> inf -> 1, underflow -> 0 -> -1
  const float e = __expf(2.0f * x);
  return 1.0f - __fdividef(2.0f, e + 1.0f);
}

__global__ __launch_bounds__(kThreads, 1)
void rnn_wmma_kernel(const float* __restrict__ X,
                     const float* __restrict__ Wc,   // (D+H, H) row-major
                     const float* __restrict__ bc,   // (H)
                     const float* __restrict__ Wo,   // (H, O)
                     const float* __restrict__ bo,   // (O)
                     float* __restrict__ out) {      // (B, O)
  __shared__ float2 sWx[kKTX * kNT * 32];            // 80 KB, B-fragment order
  __shared__ float2 sWh[kKTH * kNT * 32];            // 100 KB, B-fragment order
  __shared__ float  sX [2][kRows * kXStride];        // 2 x 16.5 KB double-buffered x_t
  __shared__ float  sH [kRows * kHStride];           // 21 KB h staging (C->A transpose)
  __shared__ float  sB [kHP];

  const int tid = threadIdx.x;
  const int w   = tid >> 5;          // wave id
  const int mw  = w >> 1;            // row-tile index (0..1)
  const int nw  = w & 1;             // N-half index (0..1): tiles [nw*5, nw*5+5)
  const int l   = tid & 31;          // lane
  const int lr  = l & 15;            // A-frag row / B-frag col within tile
  const int lh  = l >> 4;            // lane half
  const int coff = lh * 2;           // A/B frag K offset for high lane half
  const int rowBase = blockIdx.x * kRows;

  // ---- stage W_x fragments: lane sl, vgpr v holds W[K=4*kt+2*(sl>>4)+v][N=nt*16+(sl&15)]
  for (int s = tid; s < kKTX * kNT * 32; s += kThreads) {
    const int f = s >> 5, sl = s & 31;
    const int kt = f / kNT, nt = f % kNT;
    const int K = 4 * kt + 2 * (sl >> 4);
    const int N = nt * 16 + (sl & 15);
    float x = 0.f, y = 0.f;
    if (N < kH) { x = Wc[K * kH + N]; y = Wc[(K + 1) * kH + N]; }
    sWx[s] = make_float2(x, y);
  }
  // ---- stage W_h fragments (rows D..D+H-1 of W_cell), zero-padded to 160x160 ----
  for (int s = tid; s < kKTH * kNT * 32; s += kThreads) {
    const int f = s >> 5, sl = s & 31;
    const int kt = f / kNT, nt = f % kNT;
    const int K = 4 * kt + 2 * (sl >> 4);
    const int N = nt * 16 + (sl & 15);
    float x = 0.f, y = 0.f;
    if (N < kH) {
      if (K < kH)     x = Wc[(kD + K) * kH + N];
      if (K + 1 < kH) y = Wc[(kD + K + 1) * kH + N];
    }
    sWh[s] = make_float2(x, y);
  }
  for (int i = tid; i < kHP; i += kThreads) sB[i] = (i < kH) ? bc[i] : 0.f;
  for (int i = tid; i < kRows * kHStride; i += kThreads) sH[i] = 0.f;   // h0 = 0

  // ---- prologue: load x_0 into buffer 0 ----
  {
#pragma unroll
    for (int i = 0; i < kXPerThr; ++i) {
      const int s = tid + i * kThreads;
      const int row = s >> 5, c4 = s & 31;
      const float4 v = *(const float4*)(X + (((size_t)(rowBase + row)) * kT + 0) * kD + c4 * 4);
      *(float4*)(&sX[0][row * kXStride + c4 * 4]) = v;
    }
  }

  const float* aXrow0 = &sX[0][(mw * 16 + lr) * kXStride];
  const float* aXrow1 = &sX[1][(mw * 16 + lr) * kXStride];
  const float* aHrow  = &sH[(mw * 16 + lr) * kHStride];
  const int crow0 = mw * 16 + lh * 8;   // C-frag: vgpr r -> batch row crow0 + r

  for (int t = 0; t < kT; ++t) {
    // issue global loads of x_{t+1} now; drain into LDS after compute
    float4 xn[kXPerThr];
    if (t + 1 < kT) {
#pragma unroll
      for (int i = 0; i < kXPerThr; ++i) {
        const int s = tid + i * kThreads;
        const int row = s >> 5, c4 = s & 31;
        xn[i] = *(const float4*)(X + (((size_t)(rowBase + row)) * kT + (t + 1)) * kD + c4 * 4);
      }
    }
    __syncthreads();   // x_t buffer + h_{t-1} visible to all waves

    const float* aXrow = (t & 1) ? aXrow1 : aXrow0;
    v8f acc[kNTW];
#pragma unroll
    for (int n = 0; n < kNTW; ++n) {
      const float bv = sB[(nw * kNTW + n) * 16 + lr];
      acc[n] = (v8f){bv, bv, bv, bv, bv, bv, bv, bv};
    }
#pragma unroll 2
    for (int k = 0; k < kKTX; ++k) {
      const float2 av = *(const float2*)(aXrow + 4 * k + coff);
      v2f a; a.x = av.x; a.y = av.y;
      const float2* wf = &sWx[(k * kNT + nw * kNTW) * 32 + l];
#pragma unroll
      for (int n = 0; n < kNTW; ++n) {
        const float2 wv = wf[n * 32];
        v2f b; b.x = wv.x; b.y = wv.y;
        acc[n] = __builtin_amdgcn_wmma_f32_16x16x4_f32(
            false, a, false, b, (short)0, acc[n], false, false);
      }
    }
#pragma unroll 2
    for (int k = 0; k < kKTH; ++k) {
      const float2 av = *(const float2*)(aHrow + 4 * k + coff);
      v2f a; a.x = av.x; a.y = av.y;
      const float2* wf = &sWh[(k * kNT + nw * kNTW) * 32 + l];
#pragma unroll
      for (int n = 0; n < kNTW; ++n) {
        const float2 wv = wf[n * 32];
        v2f b; b.x = wv.x; b.y = wv.y;
        acc[n] = __builtin_amdgcn_wmma_f32_16x16x4_f32(
            false, a, false, b, (short)0, acc[n], false, false);
      }
    }
    // h = tanh(pre-activation)
#pragma unroll
    for (int n = 0; n < kNTW; ++n)
#pragma unroll
      for (int i = 0; i < 8; ++i)
        acc[n][i] = fast_tanh(acc[n][i]);

    // drain the x_{t+1} prefetch into the other LDS buffer (loadcnt wait
    // overlapped with the WMMA/tanh work above)
    if (t + 1 < kT) {
      float* nxt = (t & 1) ? sX[0] : sX[1];
#pragma unroll
      for (int i = 0; i < kXPerThr; ++i) {
        const int s = tid + i * kThreads;
        const int row = s >> 5, c4 = s & 31;
        *(float4*)(&nxt[row * kXStride + c4 * 4]) = xn[i];
      }
    }
    __syncthreads();   // all waves done reading sX[t&1] / old sH
#pragma unroll
    for (int n = 0; n < kNTW; ++n) {
      const int col = (nw * kNTW + n) * 16 + lr;
#pragma unroll
      for (int r = 0; r < 8; ++r)
        sH[(crow0 + r) * kHStride + col] = acc[n][r];
    }
    __syncthreads();   // new h visible
  }

  // classifier head: logits = h_T @ W_out + b_out
  for (int p = tid; p < kRows * kO; p += kThreads) {
    const int r = p / kO, o = p % kO;
    float sum = bo[o];
    const float* hrow = &sH[r * kHStride];
    for (int k = 0; k < kH; ++k) sum += hrow[k] * Wo[k * kO + o];
    out[(size_t)(rowBase + r) * kO + o] = sum;
  }
}

extern "C" void kernel_launch(void* const* d_in, const int* in_sizes, int n_in,
                              void* d_out, int out_size, void* d_ws, size_t ws_size,
                              hipStream_t stream) {
  const float* X  = (const float*)d_in[0];
  const float* Wc = (const float*)d_in[1];
  const float* bc = (const float*)d_in[2];
  const float* Wo = (const float*)d_in[3];
  const float* bo = (const float*)d_in[4];
  float* out = (float*)d_out;
  (void)in_sizes; (void)n_in; (void)out_size; (void)d_ws; (void)ws_size;
  dim3 grid(kB / kRows), block(kThreads);
  hipLaunchKernelGGL(rnn_wmma_kernel, grid, block, 0, stream, X, Wc, bc, Wo, bo, out);
}